// PositionalEncoder_41188736369188
// MI455X (gfx1250) — compile-verified
//
#include <hip/hip_runtime.h>

// Elementwise fused scale+broadcast-add:
//   out[b,t,h,w,s] = x[b,t,h,w,s] * 16.0f + pe[t,s]
// x: (16, 12, 32, 32, 256) fp32, pe: (12, 256) fp32.
// Bandwidth-bound (0.25 FLOP/byte, ~403 MB traffic -> ~17us at 23.3 TB/s).
// CDNA5 path: async global->LDS staging of pe (ASYNCcnt), NT b128 streaming.
//
// Key geometry fact: the pe broadcast pattern repeats every
// T*H*W*S/4 = 786,432 vec4s. We launch with grid-stride = 2x that period,
// so each thread's pe vector is LOOP-INVARIANT: one ds_load_b128 up front,
// then a pure load->fma->store steady state.

typedef float v4f __attribute__((ext_vector_type(4)));

#define T_MAX    12
#define S_MODEL  256
#define PE_VEC4  ((T_MAX * S_MODEL) / 4)     // 768 float4s = 12 KB
#define PE_PERIOD 786432u                    // T*H*W*S/4 vec4s

__global__ __launch_bounds__(256) void pe_encoder_kernel(
    const float* __restrict__ x,
    const float* __restrict__ pe,
    float* __restrict__ out,
    unsigned n4)   // number of float4 elements
{
    __shared__ __align__(16) float sh_pe[T_MAX * S_MODEL];

    // ---- Stage pe (12 KB) into LDS: exactly 3 async b128 copies per lane
    // (256 lanes x 3 = 768 chunks), fully unrolled, one ASYNCcnt wait.
    {
        const v4f* __restrict__ peg = (const v4f*)pe;
        const unsigned tid = threadIdx.x;
#pragma unroll
        for (unsigned k = 0; k < 3; ++k) {
            const unsigned idx = tid + 256u * k;
            const unsigned lds_addr =
                (unsigned)(unsigned long long)(sh_pe + 4u * idx);
            const v4f* gaddr = peg + idx;
            asm volatile("global_load_async_to_lds_b128 %0, %1, off"
                         :
                         : "v"(lds_addr), "v"(gaddr)
                         : "memory");
        }
#if __has_builtin(__builtin_amdgcn_s_wait_asynccnt)
        __builtin_amdgcn_s_wait_asynccnt(0);
#else
        asm volatile("s_wait_asynccnt 0" ::: "memory");
#endif
        __syncthreads();
    }

    const v4f* __restrict__ xv = (const v4f*)x;
    v4f* __restrict__       ov = (v4f*)out;

    const unsigned stride = gridDim.x * 256u;
    unsigned i = blockIdx.x * 256u + threadIdx.x;

    if ((stride % PE_PERIOD) == 0u) {
        // pe index is invariant across all of this thread's iterations:
        // stride is a multiple of the broadcast period, so neither
        // t = (i>>16)%12 nor s4 = i&63 ever changes.
        const unsigned pa = ((i >> 16) % 12u) * 64u + (i & 63u);
        const v4f pv = *(const v4f*)(sh_pe + 4u * pa);   // single ds_load_b128

        // Steady state: 4 independent NT b128 load/store streams per thread.
        for (; i + 3u * stride < n4; i += 4u * stride) {
            const unsigned i1 = i + stride;
            const unsigned i2 = i + 2u * stride;
            const unsigned i3 = i + 3u * stride;

            v4f a = __builtin_nontemporal_load(xv + i);
            v4f b = __builtin_nontemporal_load(xv + i1);
            v4f c = __builtin_nontemporal_load(xv + i2);
            v4f d = __builtin_nontemporal_load(xv + i3);

            a = a * 16.0f + pv;
            b = b * 16.0f + pv;
            c = c * 16.0f + pv;
            d = d * 16.0f + pv;

            __builtin_nontemporal_store(a, ov + i);
            __builtin_nontemporal_store(b, ov + i1);
            __builtin_nontemporal_store(c, ov + i2);
            __builtin_nontemporal_store(d, ov + i3);
        }
        for (; i < n4; i += stride) {
            v4f a = __builtin_nontemporal_load(xv + i);
            a = a * 16.0f + pv;
            __builtin_nontemporal_store(a, ov + i);
        }
    } else {
        // Generic fallback: recompute pe index per iteration.
        for (; i < n4; i += stride) {
            v4f a = __builtin_nontemporal_load(xv + i);
            const unsigned pa = ((i >> 16) % 12u) * 64u + (i & 63u);
            const v4f pva = *(const v4f*)(sh_pe + 4u * pa);
            a = a * 16.0f + pva;
            __builtin_nontemporal_store(a, ov + i);
        }
    }
}

extern "C" void kernel_launch(void* const* d_in, const int* in_sizes, int n_in,
                              void* d_out, int out_size, void* d_ws, size_t ws_size,
                              hipStream_t stream) {
    const float* x  = (const float*)d_in[0];   // 16*12*32*32*256 fp32
    const float* pe = (const float*)d_in[1];   // 12*256 fp32
    float* out      = (float*)d_out;

    const unsigned n4 = (unsigned)(in_sizes[0] / 4);   // 12,582,912 vec4s

    // 6144 blocks x 256 threads: stride = 1,572,864 vec4s = 2x pe period
    // (hoisted-pe fast path taken), n4 = 8 * stride -> exactly 8 vec4s
    // per thread = 2 unrolled-x4 iterations. 8 wave32 waves per block.
    unsigned blocks = 6144u;
    const unsigned needed = (n4 + 255u) / 256u;
    if (blocks > needed) blocks = needed;

    pe_encoder_kernel<<<blocks, 256, 0, stream>>>(x, pe, out, n4);
}